// SGLModel_22316650070691
// MI455X (gfx1250) — compile-verified
//
#include <hip/hip_runtime.h>
#include <stdint.h>

// ---------------------------------------------------------------------------
// SGL / LightGCN propagation: 3 x SpMM (COO, nnz=6M, D=64 fp32) + layer mean.
// Memory/atomic bound: embedding table (128MB) is L2-resident (192MB L2).
// CDNA5 paths: wave32 half-wave-per-edge layout, async global->LDS edge
// staging (ASYNCcnt double buffer), global_prefetch_b8, no-return f32 atomics.
// ---------------------------------------------------------------------------

#define TILE   1024          // edges staged per LDS tile (multiple of 4)
#define BLOCK  256           // 8 waves (wave32)
#define WAVES  8

// Low 32 bits of a flat shared-aperture pointer == LDS byte offset (ISA 10.2).
__device__ __forceinline__ uint32_t lds_off(const void* p) {
  return (uint32_t)(uintptr_t)p;
}

__device__ __forceinline__ void async_b128(void* lds_dst, const void* g) {
  asm volatile("global_load_async_to_lds_b128 %0, %1, off"
               :: "v"(lds_off(lds_dst)), "v"((uint64_t)(uintptr_t)g)
               : "memory");
}
__device__ __forceinline__ void async_b32(void* lds_dst, const void* g) {
  asm volatile("global_load_async_to_lds_b32 %0, %1, off"
               :: "v"(lds_off(lds_dst)), "v"((uint64_t)(uintptr_t)g)
               : "memory");
}
__device__ __forceinline__ void wait_async0() {
  asm volatile("s_wait_asynccnt 0x0" ::: "memory");
}

// Stage n (<=TILE) 4-byte elements of g[base..base+n) into lbuf via async DMA.
// Hot path: one b128 per thread (n is a multiple of 4 for this problem).
template <typename T>
__device__ __forceinline__ void stage(T* lbuf, const T* __restrict__ g,
                                      int base, int n, int tid) {
  int idx = tid * 4;
  if (idx + 4 <= n) {
    async_b128(&lbuf[idx], g + base + idx);
  } else {
    for (int k = idx; k < n; ++k) async_b32(&lbuf[k], g + base + k);
  }
}

// ---------------------------------------------------------------------------
// Edge-parallel scatter SpMM: embOut[row[e], :] += val[e] * embIn[col[e], :]
// Two edges per wave per iteration: each 16-lane half owns one edge, lane
// sl in the half owns features [4*sl, 4*sl+3] (float4 => 256B/row/half-wave).
// Edge tiles double-buffered in LDS via async loads.
// ---------------------------------------------------------------------------
__global__ __launch_bounds__(BLOCK)
void spmm_scatter(const float* __restrict__ embIn,
                  float*       __restrict__ embOut,
                  const int*   __restrict__ rowIdx,
                  const int*   __restrict__ colIdx,
                  const float* __restrict__ edgeVal,
                  int nnz, int nTiles) {
  __shared__ int   s_row[2][TILE];
  __shared__ int   s_col[2][TILE];
  __shared__ float s_val[2][TILE];

  const int tid  = threadIdx.x;
  const int lane = tid & 31;
  const int wave = tid >> 5;
  const int sub  = lane >> 4;      // which edge of the pair this half handles
  const int sl   = lane & 15;      // lane within the 16-lane half
  const int stride = gridDim.x;

  int t0 = blockIdx.x;
  if (t0 >= nTiles) return;

  { // preload first tile
    int base = t0 * TILE;
    int n = nnz - base; if (n > TILE) n = TILE;
    stage(s_row[0], rowIdx, base, n, tid);
    stage(s_col[0], colIdx, base, n, tid);
    stage(s_val[0], edgeVal, base, n, tid);
    wait_async0();
  }
  __syncthreads();

  int cur = 0;
  for (int t = t0; t < nTiles; t += stride) {
    int n = nnz - t * TILE; if (n > TILE) n = TILE;

    // kick off async copy of the NEXT tile into the other buffer
    int tn = t + stride;
    if (tn < nTiles) {
      int base2 = tn * TILE;
      int n2 = nnz - base2; if (n2 > TILE) n2 = TILE;
      stage(s_row[cur ^ 1], rowIdx, base2, n2, tid);
      stage(s_col[cur ^ 1], colIdx, base2, n2, tid);
      stage(s_val[cur ^ 1], edgeVal, base2, n2, tid);
    }

    // process current tile: two edges per wave per iteration
    for (int p = wave * 2; p < n; p += 2 * WAVES) {
      int i = p + sub;                       // this half-wave's edge
      if (i < n) {
        int   r = s_row[cur][i];
        int   c = s_col[cur][i];
        float v = s_val[cur][i];

        // look-ahead prefetch of the next gather row (global_prefetch_b8)
        int ip = i + 2 * WAVES;
        if (ip < n) {
          int cn = s_col[cur][ip];
          __builtin_prefetch(embIn + (size_t)cn * 64 + sl * 4, 0, 0);
        }

        const float4* src = (const float4*)(embIn + (size_t)c * 64);
        float4 x = src[sl];                  // global_load_b128, coalesced
        float* dst = embOut + (size_t)r * 64 + sl * 4;
        unsafeAtomicAdd(dst,     v * x.x);   // global_atomic_add_f32 no-return
        unsafeAtomicAdd(dst + 1, v * x.y);
        unsafeAtomicAdd(dst + 2, v * x.z);
        unsafeAtomicAdd(dst + 3, v * x.w);
      }
    }

    wait_async0();      // next-tile DMA must be in LDS before buffer swap
    __syncthreads();    // all waves done with this tile
    cur ^= 1;
  }
}

// acc[i] = cur[i] = concat(user_emb, item_emb)[i]
__global__ void init_concat(const float4* __restrict__ ue,
                            const float4* __restrict__ ie,
                            float4* __restrict__ cur,
                            float4* __restrict__ acc,
                            long uq, long total) {
  long i = blockIdx.x * (long)blockDim.x + threadIdx.x;
  long stride = (long)gridDim.x * blockDim.x;
  for (; i < total; i += stride) {
    float4 v = (i < uq) ? ue[i] : ie[i - uq];
    cur[i] = v;
    acc[i] = v;
  }
}

__global__ void zero_f4(float4* __restrict__ p, long total) {
  long i = blockIdx.x * (long)blockDim.x + threadIdx.x;
  long stride = (long)gridDim.x * blockDim.x;
  float4 z = {0.f, 0.f, 0.f, 0.f};
  for (; i < total; i += stride) p[i] = z;
}

// acc = (acc + nxt) * s ; optionally zero zbuf (the retired ping-pong buffer)
__global__ void add_scaled_f4(float4* __restrict__ acc,
                              const float4* __restrict__ nxt,
                              float4* __restrict__ zbuf,   // may be null
                              long total, float s) {
  long i = blockIdx.x * (long)blockDim.x + threadIdx.x;
  long stride = (long)gridDim.x * blockDim.x;
  float4 z = {0.f, 0.f, 0.f, 0.f};
  for (; i < total; i += stride) {
    float4 a = acc[i], b = nxt[i];
    a.x = (a.x + b.x) * s;
    a.y = (a.y + b.y) * s;
    a.z = (a.z + b.z) * s;
    a.w = (a.w + b.w) * s;
    acc[i] = a;
    if (zbuf) zbuf[i] = z;       // pre-zero next layer's output buffer
  }
}

extern "C" void kernel_launch(void* const* d_in, const int* in_sizes, int n_in,
                              void* d_out, int out_size, void* d_ws, size_t ws_size,
                              hipStream_t stream) {
  const float* user_emb = (const float*)d_in[0];   // [U*64]
  const float* item_emb = (const float*)d_in[1];   // [I*64]
  const float* adj_val  = (const float*)d_in[2];   // [nnz]
  const int*   adj_row  = (const int*)  d_in[3];   // [nnz]
  const int*   adj_col  = (const int*)  d_in[4];   // [nnz]
  // d_in[5] = num_layers (device scalar); fixed L=3 per the reference module.
  const int L = 3;

  const long nFloats = (long)in_sizes[0] + (long)in_sizes[1];  // N * 64
  const long uq      = (long)in_sizes[0] / 4;                  // user float4s
  const long total4  = nFloats / 4;
  const int  nnz     = in_sizes[2];

  float* acc  = (float*)d_out;         // running layer sum -> final output
  float* bufA = (float*)d_ws;          // ping
  float* bufB = bufA + nFloats;        // pong  (ws needs 2*nFloats*4 = 256 MB)

  const int EW_GRID = 2048;
  init_concat<<<EW_GRID, BLOCK, 0, stream>>>(
      (const float4*)user_emb, (const float4*)item_emb,
      (float4*)bufA, (float4*)acc, uq, total4);
  zero_f4<<<EW_GRID, BLOCK, 0, stream>>>((float4*)bufB, total4);

  const int nTiles = (nnz + TILE - 1) / TILE;
  int sgrid = nTiles < 1536 ? nTiles : 1536;   // persistent blocks -> DMA overlap

  float* cur = bufA;
  float* nxt = bufB;
  for (int l = 0; l < L; ++l) {
    spmm_scatter<<<sgrid, BLOCK, 0, stream>>>(cur, nxt, adj_row, adj_col,
                                              adj_val, nnz, nTiles);
    bool last = (l == L - 1);
    float s = last ? (1.0f / (float)(L + 1)) : 1.0f;
    // after spmm, `cur` is dead: zero it so it can be the next layer's output
    add_scaled_f4<<<EW_GRID, BLOCK, 0, stream>>>(
        (float4*)acc, (const float4*)nxt, last ? (float4*)nullptr : (float4*)cur,
        total4, s);
    float* tmp = cur; cur = nxt; nxt = tmp;
  }
}